// FpsInferModel_89386859364997
// MI455X (gfx1250) — compile-verified
//
#include <hip/hip_runtime.h>

#define NT 1024           // threads per block = 32 waves (wave32)
#define MAX_LDS_N 32768   // dist[] elements held in LDS (128 KB)

// Max-reduce a 64-bit key across the 32 lanes of a wave (wave32).
__device__ __forceinline__ unsigned long long wave_max_u64(unsigned long long k) {
#pragma unroll
  for (int off = 16; off >= 1; off >>= 1) {
    unsigned int lo = (unsigned int)k;
    unsigned int hi = (unsigned int)(k >> 32);
    unsigned int olo = (unsigned int)__shfl_xor((int)lo, off, 32);
    unsigned int ohi = (unsigned int)__shfl_xor((int)hi, off, 32);
    unsigned long long o = (((unsigned long long)ohi) << 32) | (unsigned long long)olo;
    if (o > k) k = o;
  }
  return k;
}

// key = (bits(dist) << 32) | (0xFFFFFFFF - idx): max-key == (max dist, min idx).
__device__ __forceinline__ unsigned long long make_key(float v, int idx) {
  return (((unsigned long long)__float_as_uint(v)) << 32) |
         (unsigned long long)(0xFFFFFFFFu - (unsigned int)idx);
}

// Squared distance in the reference's exact fp32 order: ((dx^2 + dy^2) + dz^2),
// _rn intrinsics block fp contraction (no fma fusing -> bit-match jnp).
__device__ __forceinline__ float sqdist(float px, float py, float pz,
                                        float lx, float ly, float lz) {
  float dx = __fsub_rn(px, lx);
  float dy = __fsub_rn(py, ly);
  float dz = __fsub_rn(pz, lz);
  return __fadd_rn(__fadd_rn(__fmul_rn(dx, dx), __fmul_rn(dy, dy)),
                   __fmul_rn(dz, dz));
}

struct F3 { float x, y, z; };

// ---------------------------------------------------------------------------
// Main kernel: one workgroup per segment, 4-point chunks per lane.
//   - dist[] LDS-resident, accessed as b128 (1 load + 1 store per 4 points)
//   - point data: 3 coalesced global_load_b128 per 48B chunk (L2-resident)
//   - argmax: strict-> per-thread scan (first occurrence), then u64-key
//     butterfly reduce intra-wave + LDS across 32 waves.
// Requires n % 4 == 0 and n <= MAX_LDS_N.
// ---------------------------------------------------------------------------
__launch_bounds__(NT, 1)
__global__ void fps_lds4_kernel(const float* __restrict__ x,
                                const int* __restrict__ ptr,
                                int* __restrict__ out,
                                int n, int m) {
  __shared__ float dist[MAX_LDS_N];
  __shared__ unsigned long long rkeys[NT / 32];
  __shared__ int bcast;

  const int t = threadIdx.x;
  const int b = blockIdx.x;
  const int base = ptr[b];
  const float* __restrict__ xs = x + (size_t)base * 3;
  const float4* __restrict__ xv = (const float4*)xs;  // 3 float4 per 4 points
  const int nchunks = n >> 2;

  // Each thread initializes only the entries it owns -> no barrier needed.
  {
    float4 inf4 = make_float4(__builtin_inff(), __builtin_inff(),
                              __builtin_inff(), __builtin_inff());
    for (int c = t; c < nchunks; c += NT) ((float4*)dist)[c] = inf4;
  }

  int last = 0;  // random_start == False -> start at local index 0
  if (t == 0) out[(size_t)b * (size_t)m] = base;  // idxs[0] = start

  for (int it = 0; it + 1 < m; ++it) {
    // Uniform (SGPR) index -> scalar broadcast load of the selected point.
    int sl = __builtin_amdgcn_readfirstlane(last);
    float lx = xs[3 * sl + 0];
    float ly = xs[3 * sl + 1];
    float lz = xs[3 * sl + 2];

    float bestv = -1.0f;  // all dist values are >= 0
    int besti = 0;
#pragma unroll 2
    for (int c = t; c < nchunks; c += NT) {
      // 4 points = 48 bytes = 3 float4 (16B-aligned since 48 = 3*16).
      float4 a  = xv[3 * c + 0];   // x0 y0 z0 x1
      float4 b4 = xv[3 * c + 1];   // y1 z1 x2 y2
      float4 cc = xv[3 * c + 2];   // z2 x3 y3 z3

      float d0 = sqdist(a.x,  a.y,  a.z,  lx, ly, lz);
      float d1 = sqdist(a.w,  b4.x, b4.y, lx, ly, lz);
      float d2 = sqdist(b4.z, b4.w, cc.x, lx, ly, lz);
      float d3 = sqdist(cc.y, cc.z, cc.w, lx, ly, lz);

      float4 dv = ((float4*)dist)[c];  // ds_load_b128
      dv.x = fminf(dv.x, d0);
      dv.y = fminf(dv.y, d1);
      dv.z = fminf(dv.z, d2);
      dv.w = fminf(dv.w, d3);
      ((float4*)dist)[c] = dv;         // ds_store_b128

      int i0 = c << 2;  // ascending within thread -> strict > keeps first max
      if (dv.x > bestv) { bestv = dv.x; besti = i0; }
      if (dv.y > bestv) { bestv = dv.y; besti = i0 + 1; }
      if (dv.z > bestv) { bestv = dv.z; besti = i0 + 2; }
      if (dv.w > bestv) { bestv = dv.w; besti = i0 + 3; }
    }

    unsigned long long key = wave_max_u64(make_key(bestv, besti));
    if ((t & 31) == 0) rkeys[t >> 5] = key;
    __syncthreads();
    if (t < NT / 32) {  // wave 0 reduces the 32 per-wave winners
      unsigned long long k2 = wave_max_u64(rkeys[t]);
      if (t == 0) {
        int nxt = (int)(0xFFFFFFFFu - (unsigned int)k2);
        bcast = nxt;
        out[(size_t)b * (size_t)m + (size_t)(it + 1)] = base + nxt;
      }
    }
    __syncthreads();
    last = bcast;
  }
}

// ---------------------------------------------------------------------------
// Per-point LDS variant for n <= MAX_LDS_N but n % 4 != 0.
// ---------------------------------------------------------------------------
__launch_bounds__(NT, 1)
__global__ void fps_lds1_kernel(const float* __restrict__ x,
                                const int* __restrict__ ptr,
                                int* __restrict__ out,
                                int n, int m) {
  __shared__ float dist[MAX_LDS_N];
  __shared__ unsigned long long rkeys[NT / 32];
  __shared__ int bcast;

  const int t = threadIdx.x;
  const int b = blockIdx.x;
  const int base = ptr[b];
  const float* __restrict__ xs = x + (size_t)base * 3;
  const F3* __restrict__ pts = (const F3*)xs;

  for (int idx = t; idx < n; idx += NT) dist[idx] = __builtin_inff();

  int last = 0;
  if (t == 0) out[(size_t)b * (size_t)m] = base;

  for (int it = 0; it + 1 < m; ++it) {
    int sl = __builtin_amdgcn_readfirstlane(last);
    float lx = xs[3 * sl + 0], ly = xs[3 * sl + 1], lz = xs[3 * sl + 2];
    float bestv = -1.0f;
    int besti = 0;
#pragma unroll 4
    for (int idx = t; idx < n; idx += NT) {
      F3 p = pts[idx];
      float d = sqdist(p.x, p.y, p.z, lx, ly, lz);
      float nd = fminf(dist[idx], d);
      dist[idx] = nd;
      if (nd > bestv) { bestv = nd; besti = idx; }
    }
    unsigned long long key = wave_max_u64(make_key(bestv, besti));
    if ((t & 31) == 0) rkeys[t >> 5] = key;
    __syncthreads();
    if (t < NT / 32) {
      unsigned long long k2 = wave_max_u64(rkeys[t]);
      if (t == 0) {
        int nxt = (int)(0xFFFFFFFFu - (unsigned int)k2);
        bcast = nxt;
        out[(size_t)b * (size_t)m + (size_t)(it + 1)] = base + nxt;
      }
    }
    __syncthreads();
    last = bcast;
  }
}

// ---------------------------------------------------------------------------
// Fallback for n > MAX_LDS_N: dist in global scratch (d_ws).
// ---------------------------------------------------------------------------
__launch_bounds__(NT, 1)
__global__ void fps_glob_kernel(const float* __restrict__ x,
                                const int* __restrict__ ptr,
                                int* __restrict__ out,
                                float* __restrict__ distg,
                                int n, int m) {
  __shared__ unsigned long long rkeys[NT / 32];
  __shared__ int bcast;

  const int t = threadIdx.x;
  const int b = blockIdx.x;
  const int base = ptr[b];
  const float* __restrict__ xs = x + (size_t)base * 3;
  const F3* __restrict__ pts = (const F3*)xs;
  float* __restrict__ dist = distg + (size_t)b * (size_t)n;

  for (int idx = t; idx < n; idx += NT) dist[idx] = __builtin_inff();

  int last = 0;
  if (t == 0) out[(size_t)b * (size_t)m] = base;

  for (int it = 0; it + 1 < m; ++it) {
    int sl = __builtin_amdgcn_readfirstlane(last);
    float lx = xs[3 * sl + 0], ly = xs[3 * sl + 1], lz = xs[3 * sl + 2];
    float bestv = -1.0f;
    int besti = 0;
#pragma unroll 2
    for (int idx = t; idx < n; idx += NT) {
      F3 p = pts[idx];
      float d = sqdist(p.x, p.y, p.z, lx, ly, lz);
      float nd = fminf(dist[idx], d);
      dist[idx] = nd;
      if (nd > bestv) { bestv = nd; besti = idx; }
    }
    unsigned long long key = wave_max_u64(make_key(bestv, besti));
    if ((t & 31) == 0) rkeys[t >> 5] = key;
    __syncthreads();
    if (t < NT / 32) {
      unsigned long long k2 = wave_max_u64(rkeys[t]);
      if (t == 0) {
        int nxt = (int)(0xFFFFFFFFu - (unsigned int)k2);
        bcast = nxt;
        out[(size_t)b * (size_t)m + (size_t)(it + 1)] = base + nxt;
      }
    }
    __syncthreads();
    last = bcast;
  }
}

extern "C" void kernel_launch(void* const* d_in, const int* in_sizes, int n_in,
                              void* d_out, int out_size, void* d_ws, size_t ws_size,
                              hipStream_t stream) {
  const float* x  = (const float*)d_in[0];   // [B*n, 3] float32
  const int* ptr  = (const int*)d_in[1];     // [B+1]  int32 segment boundaries
  // d_in[2] = ratio (implied by out_size), d_in[3] = random_start (False)
  int* out = (int*)d_out;                    // [B*m] int32 global indices

  int B = in_sizes[1] - 1;
  if (B <= 0) return;
  int n = in_sizes[0] / (B * 3);             // equal-size segments, D == 3
  int m = out_size / B;                      // m = ceil(ratio * n)
  if (n <= 0 || m <= 0) return;

  if (n <= MAX_LDS_N && (n & 3) == 0) {
    hipLaunchKernelGGL(fps_lds4_kernel, dim3(B), dim3(NT), 0, stream,
                       x, ptr, out, n, m);
  } else if (n <= MAX_LDS_N) {
    hipLaunchKernelGGL(fps_lds1_kernel, dim3(B), dim3(NT), 0, stream,
                       x, ptr, out, n, m);
  } else {
    hipLaunchKernelGGL(fps_glob_kernel, dim3(B), dim3(NT), 0, stream,
                       x, ptr, out, (float*)d_ws, n, m);
  }
}